// BasisSignalLayer_10763188043901
// MI455X (gfx1250) — compile-verified
//
#include <hip/hip_runtime.h>

// ---------------------------------------------------------------------------
// BasisSignalLayer: out = overlap_and_add(weight @ W^T, hop=32)
//   weight: (8, 16000, 512) f32   W: (64, 512) f32   out: (8, 512032) f32
// GEMM: M=128000 (frames), N=64 (frame_len), K=512 (basis), via
// V_WMMA_F32_16X16X4_F32. Overlap-add fused via global_atomic_add_f32
// (each out element gets <=2 adds; fp add is commutative -> deterministic).
// ---------------------------------------------------------------------------

typedef __attribute__((ext_vector_type(2))) float v2f;
typedef __attribute__((ext_vector_type(4))) float v4f;
typedef __attribute__((ext_vector_type(8))) float v8f;

#define N_BASIS   512
#define FRAME_LEN 64
#define STEP      32
#define BATCH     8
#define FRAMES    16000
#define OUT_PER_B (STEP * (FRAMES - 1) + FRAME_LEN)   // 512032

#define KPHASE            128                  // K columns staged in LDS per phase
#define NPHASES           (N_BASIS / KPHASE)   // 4
#define LDS_STRIDE        (KPHASE + 4)         // 132 dwords; 132 % 64 == 4 -> bank-conflict-free b64 frag reads
#define FRAMES_PER_BLOCK  128                  // 8 waves x 16 frames
#define BLOCKS_PER_BATCH  (FRAMES / FRAMES_PER_BLOCK)  // 125

#if defined(__HIP_DEVICE_COMPILE__)
#define FATOMIC_ADD(p, v) unsafeAtomicAdd((p), (v))   // -> global_atomic_add_f32
#else
#define FATOMIC_ADD(p, v) atomicAdd((p), (v))
#endif

__global__ __launch_bounds__(256) void bsl_zero_out(float4* __restrict__ out, int n4) {
    int i = blockIdx.x * blockDim.x + threadIdx.x;
    if (i < n4) out[i] = make_float4(0.f, 0.f, 0.f, 0.f);
}

__global__ __launch_bounds__(256) void bsl_gemm_olap(const float* __restrict__ weight,
                                                     const float* __restrict__ basis,
                                                     float* __restrict__ out) {
    __shared__ float sB[FRAME_LEN * LDS_STRIDE];   // 64 x 132 dwords = 33 KB

    const int tid  = threadIdx.x;
    const int lane = tid & 31;
    const int wave = tid >> 5;
    const int l16  = lane & 15;
    const int half = lane >> 4;          // 0: K pair {0,1}, 1: K pair {2,3}

    const int blk   = blockIdx.x;
    const int batch = blk / BLOCKS_PER_BATCH;
    const int fo    = (blk % BLOCKS_PER_BATCH) * FRAMES_PER_BLOCK;

    // This lane's A row (frame) pointer: A-fragment M = l16 within the wave tile.
    const float* __restrict__ aptr =
        weight + ((size_t)batch * FRAMES + (size_t)(fo + wave * 16 + l16)) * N_BASIS;

    v8f acc[4];
    #pragma unroll
    for (int nt = 0; nt < 4; ++nt) acc[nt] = v8f{0.f, 0.f, 0.f, 0.f, 0.f, 0.f, 0.f, 0.f};

    for (int ph = 0; ph < NPHASES; ++ph) {
        const int kbase = ph * KPHASE;

        __syncthreads();   // previous phase's LDS reads done before overwrite
        // Stage B phase: 64 rows x 128 cols, coalesced float4 loads.
        #pragma unroll
        for (int it = 0; it < (FRAME_LEN * KPHASE / 4) / 256; ++it) {   // 8 iters
            int idx = tid + it * 256;
            int row = idx >> 5;      // 32 float4 per row
            int c4  = idx & 31;
            v4f v = *(const v4f*)(basis + (size_t)row * N_BASIS + kbase + c4 * 4);
            *(v4f*)(&sB[row * LDS_STRIDE + c4 * 4]) = v;
        }
        __syncthreads();

        #pragma unroll 8
        for (int kb = 0; kb < KPHASE / 4; ++kb) {   // 32 K-steps of 4
            // A 16x4 f32 fragment: VGPR0/1 = A[M=l16][K = 2*half + {0,1}]
            v2f a = *(const v2f*)(aptr + kbase + kb * 4 + 2 * half);
            #pragma unroll
            for (int nt = 0; nt < 4; ++nt) {
                // B 4x16 f32 fragment: VGPR0/1 = W[N = nt*16+l16][K = 2*half + {0,1}]
                v2f b = *(const v2f*)(&sB[(nt * 16 + l16) * LDS_STRIDE + kb * 4 + 2 * half]);
                acc[nt] = __builtin_amdgcn_wmma_f32_16x16x4_f32(
                    /*neg_a=*/false, a, /*neg_b=*/false, b,
                    /*c_mod=*/(short)0, acc[nt],
                    /*reuse_a=*/false, /*reuse_b=*/false);
            }
        }
    }

    // Epilogue: D VGPR i holds (M = i + 8*half, N = l16).
    // out[batch, frame*32 + col] += D[m][n];  frame = fo + wave*16 + m, col = nt*16 + n.
    const size_t outBase = (size_t)batch * OUT_PER_B +
                           (size_t)(fo + wave * 16 + 8 * half) * STEP + l16;
    #pragma unroll
    for (int nt = 0; nt < 4; ++nt) {
        float* o = out + outBase + nt * 16;
        #pragma unroll
        for (int i = 0; i < 8; ++i) {
            FATOMIC_ADD(o + (size_t)i * STEP, acc[nt][i]);
        }
    }
}

extern "C" void kernel_launch(void* const* d_in, const int* in_sizes, int n_in,
                              void* d_out, int out_size, void* d_ws, size_t ws_size,
                              hipStream_t stream) {
    const float* weight = (const float*)d_in[0];   // (8, 16000, 512) f32
    const float* basis  = (const float*)d_in[1];   // (64, 512) f32
    float* out = (float*)d_out;                    // (8, 512032) f32

    const int n4 = (BATCH * OUT_PER_B) / 4;        // 1,024,064 float4s
    bsl_zero_out<<<(n4 + 255) / 256, 256, 0, stream>>>((float4*)out, n4);

    bsl_gemm_olap<<<BATCH * BLOCKS_PER_BATCH, 256, 0, stream>>>(weight, basis, out);
}